// FocalLossWithMask_64441689309323
// MI455X (gfx1250) — compile-verified
//
#include <hip/hip_runtime.h>
#include <stdint.h>

// Problem geometry (fixed by the reference).
#define B_ROWS 64
#define N_COLS 131072
#define NBINS  1024
#define THREADS 256
#define VEC 4                                       // floats per lane per stage (B128)
#define STAGE_ELEMS (THREADS * VEC)                 // 1024 floats / stage
#define SLABS_PER_ROW 32
#define ELEMS_PER_BLOCK (N_COLS / SLABS_PER_ROW)    // 4096 floats / block
#define STAGES (ELEMS_PER_BLOCK / STAGE_ELEMS)      // 4 double-buffered stages

// ---------------------------------------------------------------------------
// Focal loss, elementwise (GAMMA=2, ALPHA=0.75), numerically stable BCE.
// ---------------------------------------------------------------------------
__device__ __forceinline__ float focal_elem(float p, float l) {
    float bce  = fmaxf(p, 0.f) - p * l + __logf(1.f + __expf(-fabsf(p)));
    float prob = 1.f / (1.f + __expf(-p));
    float pt   = l * prob + (1.f - l) * (1.f - prob);
    float m    = 1.000001f - pt;
    float af   = l * 0.75f + (1.f - l) * 0.25f;
    return bce * m * m * af;
}

// gfx1250 async global->LDS copy (ASYNCcnt-tracked). One wave instruction
// moves 32 lanes x 16B = 512B. LDS dest address is the low 32 bits of the
// generic pointer (LDS aperture maps addr[31:0] -> LDS offset).
__device__ __forceinline__ void async_load_b128(unsigned int lds_addr, const void* gaddr) {
    asm volatile("global_load_async_to_lds_b128 %0, %1, off"
                 :: "v"(lds_addr), "v"(gaddr)
                 : "memory");
}
__device__ __forceinline__ unsigned int lds_addr_of(const void* p) {
    return (unsigned int)(uintptr_t)p;
}

// ---------------------------------------------------------------------------
// Kernel 0: zero the workspace counters/histograms (capture-safe init).
// ---------------------------------------------------------------------------
__global__ void k_zero(unsigned int* __restrict__ p, int n) {
    int i = blockIdx.x * blockDim.x + threadIdx.x;
    if (i < n) p[i] = 0u;
}

// ---------------------------------------------------------------------------
// Kernel 1: streaming pass 1 — per-row positive count + negative-loss
// histogram (1024 bins over the float bit pattern; monotone for loss >= 0).
// Double-buffered async LDS staging pipeline.
// ---------------------------------------------------------------------------
__global__ void k_hist(const float* __restrict__ pred,
                       const float* __restrict__ label,
                       unsigned int* __restrict__ rowHist,
                       unsigned int* __restrict__ rowNumPos) {
    __shared__ float sP[2][STAGE_ELEMS];
    __shared__ float sL[2][STAGE_ELEMS];
    __shared__ unsigned int hist[NBINS];
    __shared__ unsigned int blkPos;

    const int tid  = threadIdx.x;
    const int row  = blockIdx.x / SLABS_PER_ROW;
    const int slab = blockIdx.x % SLABS_PER_ROW;
    const float* pr = pred  + (size_t)row * N_COLS + (size_t)slab * ELEMS_PER_BLOCK;
    const float* lr = label + (size_t)row * N_COLS + (size_t)slab * ELEMS_PER_BLOCK;

    for (int i = tid; i < NBINS; i += THREADS) hist[i] = 0u;
    if (tid == 0) blkPos = 0u;
    __syncthreads();

    const int base = tid * VEC;
    const unsigned int lpA = lds_addr_of(&sP[0][base]);
    const unsigned int lpB = lds_addr_of(&sP[1][base]);
    const unsigned int llA = lds_addr_of(&sL[0][base]);
    const unsigned int llB = lds_addr_of(&sL[1][base]);

    // Prologue: stage 0 in flight.
    async_load_b128(lpA, pr + base);
    async_load_b128(llA, lr + base);

    unsigned int npos = 0u;
#pragma unroll
    for (int s = 0; s < STAGES; ++s) {
        const int buf = s & 1;
        if (s + 1 < STAGES) {
            // Prefetch next stage into the other buffer, then wait until only
            // those 2 ops remain outstanding (loads complete in order).
            async_load_b128(buf ? lpA : lpB, pr + (s + 1) * STAGE_ELEMS + base);
            async_load_b128(buf ? llA : llB, lr + (s + 1) * STAGE_ELEMS + base);
            asm volatile("s_wait_asynccnt 0x2" ::: "memory");
        } else {
            asm volatile("s_wait_asynccnt 0x0" ::: "memory");
        }
#pragma unroll
        for (int j = 0; j < VEC; ++j) {
            float p = sP[buf][base + j];
            float l = sL[buf][base + j];
            float loss = focal_elem(p, l);
            if (l > 0.f) {
                ++npos;
            } else {
                // loss >= 0 -> sign bit 0 -> bits>>21 in [0,1023], monotone.
                atomicAdd(&hist[__float_as_uint(loss) >> 21], 1u);
            }
        }
    }

    atomicAdd(&blkPos, npos);
    __syncthreads();
    if (tid == 0 && blkPos) atomicAdd(&rowNumPos[row], blkPos);
    for (int i = tid; i < NBINS; i += THREADS)
        if (hist[i]) atomicAdd(&rowHist[(size_t)row * NBINS + i], hist[i]);
}

// ---------------------------------------------------------------------------
// Kernel 2: per-row threshold bin via suffix scan of the histogram.
// k = 3*num_pos (or 3 if num_pos == 0). If k exceeds the negative count the
// scan never trips and T stays 0 (keep everything) — the reference's regime.
// ---------------------------------------------------------------------------
__global__ void k_thresh(const unsigned int* __restrict__ rowHist,
                         const unsigned int* __restrict__ rowNumPos,
                         unsigned int* __restrict__ rowThresh) {
    const int row = blockIdx.x;
    if (threadIdx.x == 0) {
        unsigned int np = rowNumPos[row];
        unsigned int k  = np ? 3u * np : 3u;
        const unsigned int* h = rowHist + (size_t)row * NBINS;
        unsigned int acc = 0u, T = 0u;
        for (int b = NBINS - 1; b >= 0; --b) {
            acc += h[b];
            if (acc >= k) { T = (unsigned int)b; break; }
        }
        rowThresh[row] = T;
    }
}

// ---------------------------------------------------------------------------
// Kernel 3: streaming pass 2 — masked sum + count (same async pipeline).
// ---------------------------------------------------------------------------
__global__ void k_sum(const float* __restrict__ pred,
                      const float* __restrict__ label,
                      const unsigned int* __restrict__ rowThresh,
                      float* __restrict__ totalSum,
                      unsigned int* __restrict__ totalCnt) {
    __shared__ float sP[2][STAGE_ELEMS];
    __shared__ float sL[2][STAGE_ELEMS];
    __shared__ float bsum;
    __shared__ unsigned int bcnt;

    const int tid  = threadIdx.x;
    const int row  = blockIdx.x / SLABS_PER_ROW;
    const int slab = blockIdx.x % SLABS_PER_ROW;
    const float* pr = pred  + (size_t)row * N_COLS + (size_t)slab * ELEMS_PER_BLOCK;
    const float* lr = label + (size_t)row * N_COLS + (size_t)slab * ELEMS_PER_BLOCK;
    const unsigned int T = rowThresh[row];

    if (tid == 0) { bsum = 0.f; bcnt = 0u; }
    __syncthreads();

    const int base = tid * VEC;
    const unsigned int lpA = lds_addr_of(&sP[0][base]);
    const unsigned int lpB = lds_addr_of(&sP[1][base]);
    const unsigned int llA = lds_addr_of(&sL[0][base]);
    const unsigned int llB = lds_addr_of(&sL[1][base]);

    async_load_b128(lpA, pr + base);
    async_load_b128(llA, lr + base);

    float lsum = 0.f;
    unsigned int lcnt = 0u;
#pragma unroll
    for (int s = 0; s < STAGES; ++s) {
        const int buf = s & 1;
        if (s + 1 < STAGES) {
            async_load_b128(buf ? lpA : lpB, pr + (s + 1) * STAGE_ELEMS + base);
            async_load_b128(buf ? llA : llB, lr + (s + 1) * STAGE_ELEMS + base);
            asm volatile("s_wait_asynccnt 0x2" ::: "memory");
        } else {
            asm volatile("s_wait_asynccnt 0x0" ::: "memory");
        }
#pragma unroll
        for (int j = 0; j < VEC; ++j) {
            float p = sP[buf][base + j];
            float l = sL[buf][base + j];
            float loss = focal_elem(p, l);
            bool rel = (l > 0.f) || ((__float_as_uint(loss) >> 21) >= T);
            if (rel) { lsum += loss; ++lcnt; }
        }
    }

    atomicAdd(&bsum, lsum);
    atomicAdd(&bcnt, lcnt);
    __syncthreads();
    if (tid == 0) {
        atomicAdd(totalSum, bsum);
        atomicAdd(totalCnt, bcnt);
    }
}

// ---------------------------------------------------------------------------
// Kernel 4: final scalar.
// ---------------------------------------------------------------------------
__global__ void k_final(const float* __restrict__ totalSum,
                        const unsigned int* __restrict__ totalCnt,
                        float* __restrict__ out) {
    float s = *totalSum;
    float d = fmaxf((float)(*totalCnt), 1.f);
    out[0] = (s != 0.f) ? s / d : 0.f;
}

// ---------------------------------------------------------------------------
// Launch.  Workspace layout (bytes):
//   [0, 262144)           rowHist   u32[64][1024]
//   [262144, 262400)      rowNumPos u32[64]
//   [262400, 262656)      rowThresh u32[64]
//   [262656, 262660)      totalSum  f32
//   [262660, 262664)      totalCnt  u32
// ---------------------------------------------------------------------------
extern "C" void kernel_launch(void* const* d_in, const int* in_sizes, int n_in,
                              void* d_out, int out_size, void* d_ws, size_t ws_size,
                              hipStream_t stream) {
    const float* pred  = (const float*)d_in[0];
    const float* label = (const float*)d_in[1];
    float* out = (float*)d_out;

    uint8_t* ws = (uint8_t*)d_ws;
    unsigned int* rowHist   = (unsigned int*)(ws);
    unsigned int* rowNumPos = (unsigned int*)(ws + 262144);
    unsigned int* rowThresh = (unsigned int*)(ws + 262400);
    float*        totalSum  = (float*)       (ws + 262656);
    unsigned int* totalCnt  = (unsigned int*)(ws + 262660);

    const int zero_words = (262664 + 3) / 4;
    k_zero<<<(zero_words + THREADS - 1) / THREADS, THREADS, 0, stream>>>(
        (unsigned int*)ws, zero_words);

    dim3 grid(B_ROWS * SLABS_PER_ROW);   // 2048 blocks, 256 threads (8 waves)
    k_hist<<<grid, THREADS, 0, stream>>>(pred, label, rowHist, rowNumPos);
    k_thresh<<<B_ROWS, 32, 0, stream>>>(rowHist, rowNumPos, rowThresh);
    k_sum<<<grid, THREADS, 0, stream>>>(pred, label, rowThresh, totalSum, totalCnt);
    k_final<<<1, 1, 0, stream>>>(totalSum, totalCnt, out);
}